// SelectiveSSM_27066883899746
// MI455X (gfx1250) — compile-verified
//
#include <hip/hip_runtime.h>
#include <hip/hip_bf16.h>
#include <math.h>

typedef float v2f __attribute__((ext_vector_type(2)));
typedef float v8f __attribute__((ext_vector_type(8)));

#define HIDDEN 1024
#define STATE 16
#define DT_RANK 64
#define NB 2
#define NS 2048
#define BSROWS (NB * NS)            // 4096 token rows
#define PROJ (2 * STATE + DT_RANK)  // 96
#define CHUNKS 64
#define CLEN (NS / CHUNKS)          // 32 steps per chunk

__device__ __forceinline__ v8f wmma_f32_k4(v2f a, v2f b, v8f c) {
  // D = A(16x4,f32) * B(4x16,f32) + C(16x16,f32)
  return __builtin_amdgcn_wmma_f32_16x16x4_f32(false, a, false, b, (short)0, c,
                                               false, false);
}

// ---------------------------------------------------------------------------
// Kernel 1: xp = x @ W_xproj, split into Bm (cols 0..15), Cm (16..31),
//           dtr (32..95). One wave per 16x16 output tile, K=1024 in steps of 4.
// ---------------------------------------------------------------------------
__global__ __launch_bounds__(128) void xproj_kernel(
    const float* __restrict__ x, const float* __restrict__ Wp,
    float* __restrict__ Bm, float* __restrict__ Cm, float* __restrict__ dtr) {
  const int lane = threadIdx.x & 31;
  const int wave = threadIdx.x >> 5;
  const int wgid = blockIdx.x * 4 + wave;          // 1536 waves total
  const int rowTile = wgid / 6;                    // 0..255
  const int colTile = wgid % 6;                    // 0..5
  const int m0 = rowTile * 16;
  const int n0 = colTile * 16;
  const int half = lane >> 4;                      // K-half (0: K=0,1  1: K=2,3)
  const int l = lane & 15;

  const float* arow = x + (m0 + l) * HIDDEN + half * 2;
  v8f acc = {};
#pragma unroll 4
  for (int k = 0; k < HIDDEN; k += 4) {
    v2f a = *(const v2f*)(arow + k);               // A[m][kk..kk+1]
    const int kk = k + half * 2;
    v2f b;
    b.x = Wp[(kk + 0) * PROJ + n0 + l];            // B[kk][n]
    b.y = Wp[(kk + 1) * PROJ + n0 + l];
    acc = wmma_f32_k4(a, b, acc);
  }
#pragma unroll
  for (int r = 0; r < 8; ++r) {
    const int row = m0 + r + half * 8;             // C layout: M=r / r+8
    const int col = n0 + l;                        // N = lane%16
    const float v = acc[r];
    if (colTile == 0)
      Bm[row * STATE + col] = v;
    else if (colTile == 1)
      Cm[row * STATE + (col - STATE)] = v;
    else
      dtr[row * DT_RANK + (col - 2 * STATE)] = v;
  }
}

// ---------------------------------------------------------------------------
// Kernel 2: dt = softplus(dtr @ W_dt + b_dt).  K=64 in steps of 4.
// ---------------------------------------------------------------------------
__global__ __launch_bounds__(128) void dt_kernel(
    const float* __restrict__ dtr, const float* __restrict__ Wdt,
    const float* __restrict__ bdt, float* __restrict__ dt) {
  const int lane = threadIdx.x & 31;
  const int wave = threadIdx.x >> 5;
  const int wgid = blockIdx.x * 4 + wave;          // 16384 waves total
  const int rowTile = wgid >> 6;                   // 0..255
  const int colTile = wgid & 63;                   // 0..63
  const int m0 = rowTile * 16;
  const int n0 = colTile * 16;
  const int half = lane >> 4;
  const int l = lane & 15;

  const float* arow = dtr + (m0 + l) * DT_RANK + half * 2;
  v8f acc = {};
#pragma unroll
  for (int k = 0; k < DT_RANK; k += 4) {
    v2f a = *(const v2f*)(arow + k);
    const int kk = k + half * 2;
    v2f b;
    b.x = Wdt[(kk + 0) * HIDDEN + n0 + l];
    b.y = Wdt[(kk + 1) * HIDDEN + n0 + l];
    acc = wmma_f32_k4(a, b, acc);
  }
  const float bias = bdt[n0 + l];
#pragma unroll
  for (int r = 0; r < 8; ++r) {
    const int row = m0 + r + half * 8;
    const float v = acc[r] + bias;
    // numerically stable softplus
    const float sp = fmaxf(v, 0.0f) + log1pf(__expf(-fabsf(v)));
    dt[row * HIDDEN + n0 + l] = sp;
  }
}

__device__ __forceinline__ void load16(const float* __restrict__ p, float* v) {
  const float4* q = (const float4*)p;  // 16B aligned (STATE-float rows)
  float4 a = q[0], b = q[1], c = q[2], d = q[3];
  v[0] = a.x;  v[1] = a.y;  v[2] = a.z;  v[3] = a.w;
  v[4] = b.x;  v[5] = b.y;  v[6] = b.z;  v[7] = b.w;
  v[8] = c.x;  v[9] = c.y;  v[10] = c.z; v[11] = c.w;
  v[12] = d.x; v[13] = d.y; v[14] = d.z; v[15] = d.w;
}

// ---------------------------------------------------------------------------
// Kernel 3: chunked scan, phase 1. Each thread owns one (b, chunk, d) channel,
// scans CLEN steps from h=0 recording h_out and the cumulative A-bar product.
// ---------------------------------------------------------------------------
__global__ __launch_bounds__(256) void scan_phase1(
    const float* __restrict__ x, const float* __restrict__ dt,
    const float* __restrict__ Bm, const float* __restrict__ A_log,
    float* __restrict__ carry_h, float* __restrict__ carry_ap) {
  const int gid = blockIdx.x * blockDim.x + threadIdx.x;  // NB*CHUNKS*HIDDEN
  const int d = gid & (HIDDEN - 1);
  const int c = (gid >> 10) & (CHUNKS - 1);
  const int b = gid >> 16;

  float A[STATE], h[STATE], Ap[STATE];
#pragma unroll
  for (int n = 0; n < STATE; ++n) {
    A[n] = -__expf(A_log[n * HIDDEN + d]);
    h[n] = 0.0f;
    Ap[n] = 1.0f;
  }
  const int s0 = c * CLEN;
  for (int s = s0; s < s0 + CLEN; ++s) {
    const int t = b * NS + s;
    const float dtv = dt[t * HIDDEN + d];
    const float bu = dtv * x[t * HIDDEN + d];
    float bm[STATE];
    load16(Bm + t * STATE, bm);
#pragma unroll
    for (int n = 0; n < STATE; ++n) {
      const float ab = __expf(A[n] * dtv);
      h[n] = fmaf(ab, h[n], bm[n] * bu);
      Ap[n] *= ab;
    }
  }
  const int base = ((b * CHUNKS + c) * STATE) * HIDDEN + d;
#pragma unroll
  for (int n = 0; n < STATE; ++n) {
    carry_h[base + n * HIDDEN] = h[n];
    carry_ap[base + n * HIDDEN] = Ap[n];
  }
}

// ---------------------------------------------------------------------------
// Kernel 4: chain chunk carries per (b,d) channel; rewrite carry_h[c] with the
// carry-IN for chunk c (in place).
// ---------------------------------------------------------------------------
__global__ __launch_bounds__(256) void combine_kernel(
    float* __restrict__ carry_h, const float* __restrict__ carry_ap) {
  const int gid = blockIdx.x * blockDim.x + threadIdx.x;  // NB*HIDDEN
  const int d = gid & (HIDDEN - 1);
  const int b = gid >> 10;
  float carry[STATE];
#pragma unroll
  for (int n = 0; n < STATE; ++n) carry[n] = 0.0f;
  for (int c = 0; c < CHUNKS; ++c) {
    const int base = ((b * CHUNKS + c) * STATE) * HIDDEN + d;
#pragma unroll
    for (int n = 0; n < STATE; ++n) {
      const int idx = base + n * HIDDEN;
      const float hout = carry_h[idx];
      const float ap = carry_ap[idx];
      carry_h[idx] = carry[n];                 // carry-in for this chunk
      carry[n] = fmaf(ap, carry[n], hout);     // carry-out
    }
  }
}

// ---------------------------------------------------------------------------
// Kernel 5: phase 3 — rescan each chunk from its true carry-in and emit
// y[t,d] = sum_n Cm[t,n] * h[n].
// ---------------------------------------------------------------------------
__global__ __launch_bounds__(256) void scan_phase3(
    const float* __restrict__ x, const float* __restrict__ dt,
    const float* __restrict__ Bm, const float* __restrict__ Cm,
    const float* __restrict__ A_log, const float* __restrict__ carry_h,
    float* __restrict__ y) {
  const int gid = blockIdx.x * blockDim.x + threadIdx.x;
  const int d = gid & (HIDDEN - 1);
  const int c = (gid >> 10) & (CHUNKS - 1);
  const int b = gid >> 16;

  float A[STATE], h[STATE];
  const int base = ((b * CHUNKS + c) * STATE) * HIDDEN + d;
#pragma unroll
  for (int n = 0; n < STATE; ++n) {
    A[n] = -__expf(A_log[n * HIDDEN + d]);
    h[n] = carry_h[base + n * HIDDEN];
  }
  const int s0 = c * CLEN;
  for (int s = s0; s < s0 + CLEN; ++s) {
    const int t = b * NS + s;
    const float dtv = dt[t * HIDDEN + d];
    const float bu = dtv * x[t * HIDDEN + d];
    float bm[STATE], cm[STATE];
    load16(Bm + t * STATE, bm);
    load16(Cm + t * STATE, cm);
    float yacc = 0.0f;
#pragma unroll
    for (int n = 0; n < STATE; ++n) {
      const float ab = __expf(A[n] * dtv);
      h[n] = fmaf(ab, h[n], bm[n] * bu);
      yacc = fmaf(cm[n], h[n], yacc);
    }
    y[t * HIDDEN + d] = yacc;
  }
}

// ---------------------------------------------------------------------------
// Workspace layout (floats):
//   [0)          Bm        : BSROWS*STATE   =   65536
//   [65536)      Cm        : BSROWS*STATE   =   65536
//   [131072)     dtr       : BSROWS*DT_RANK =  262144
//   [393216)     dt        : BSROWS*HIDDEN  = 4194304
//   [4587520)    carry_h   : NB*CHUNKS*STATE*HIDDEN = 2097152
//   [6684672)    carry_ap  : 2097152
//   total ~35.1 MB
// ---------------------------------------------------------------------------
extern "C" void kernel_launch(void* const* d_in, const int* in_sizes, int n_in,
                              void* d_out, int out_size, void* d_ws,
                              size_t ws_size, hipStream_t stream) {
  const float* x = (const float*)d_in[0];
  const float* Wp = (const float*)d_in[1];
  const float* Wdt = (const float*)d_in[2];
  const float* bdt = (const float*)d_in[3];
  const float* A_log = (const float*)d_in[4];
  float* y = (float*)d_out;

  float* ws = (float*)d_ws;
  float* Bm = ws;
  float* Cm = ws + 65536;
  float* dtr = ws + 131072;
  float* dt = ws + 393216;
  float* carry_h = ws + 4587520;
  float* carry_ap = ws + 6684672;

  // GEMM 1: 256 row tiles x 6 col tiles = 1536 waves, 4 waves/block
  xproj_kernel<<<384, 128, 0, stream>>>(x, Wp, Bm, Cm, dtr);
  // GEMM 2: 256 x 64 tiles = 16384 waves
  dt_kernel<<<4096, 128, 0, stream>>>(dtr, Wdt, bdt, dt);
  // Chunked scan
  scan_phase1<<<(NB * CHUNKS * HIDDEN) / 256, 256, 0, stream>>>(
      x, dt, Bm, A_log, carry_h, carry_ap);
  combine_kernel<<<(NB * HIDDEN) / 256, 256, 0, stream>>>(carry_h, carry_ap);
  scan_phase3<<<(NB * CHUNKS * HIDDEN) / 256, 256, 0, stream>>>(
      x, dt, Bm, Cm, A_log, carry_h, y);
}